// Attention_12403865551639
// MI455X (gfx1250) — compile-verified
//
#include <hip/hip_runtime.h>
#include <hip/hip_bf16.h>

// Problem constants (from reference setup_inputs)
#define BATCH 4
#define TQ    512
#define TK    512
#define QDIM  1024   // == KDIM, GEMM reduction depth
#define CDIM  256

typedef float v2f __attribute__((ext_vector_type(2)));
typedef float v8f __attribute__((ext_vector_type(8)));

// Hardware tanh (CDNA5 TRANS op). Much cheaper than ocml polynomial tanhf.
__device__ __forceinline__ float ftanh(float x) {
  float r;
  asm("v_tanh_f32 %0, %1" : "=v"(r) : "v"(x));
  return r;
}

// ---------------------------------------------------------------------------
// Stage 1: Y[M,CDIM] = X[M,QDIM] * W[QDIM,CDIM] with V_WMMA_F32_16X16X4_F32.
// Block = 8 waves = 128 M-rows x 16 N-cols. The 16-wide W panel is staged in
// LDS *transposed* (Wt[n][k], stride 1028 words) so each K-step is exactly:
//   1x global_load_b64 (A pair, immediate offset)
//   1x ds_load (B pair; compiler fuses pairs into ds_load_2addr_b64)
//   1x v_wmma_f32_16x16x4_f32
// All dims are compile-time constants -> no remainder loop, no guard branches.
//
// f32 A (16x4) layout (ISA 7.12.2): lane 0-15 -> M=lane, VGPR0=K0,VGPR1=K1;
//                                   lane 16-31 -> M=lane-16, VGPR0=K2,VGPR1=K3.
// f32 B (4x16): mirrored (N = lane&15, K pair from lane>>4).
// f32 C/D (16x16): vgpr r, lane l -> M = r + 8*(l>>4), N = l&15.
// ---------------------------------------------------------------------------
#define WSTR 1028   // padded LDS row stride (floats): b64 reads hit all 64 banks

__global__ __launch_bounds__(256)
void proj_gemm_wmma(const float* __restrict__ X, const float* __restrict__ W,
                    float* __restrict__ Y) {
  __shared__ __align__(16) float ldsB[16 * WSTR];   // Wt[n][k]

  const int tid  = threadIdx.x;
  const int lane = tid & 31;
  const int wv   = tid >> 5;          // wave id 0..7
  const int half = lane >> 4;         // 0 or 1
  const int l16  = lane & 15;
  const int kg   = half * 2;          // K offset within the group of 4

  const int m0 = blockIdx.x * 128;
  const int n0 = blockIdx.y * 16;

  // Stage W[:, n0:n0+16] into LDS transposed: ldsB[n][k] = W[k][n0+n].
  // float4 global loads + register transpose; scattered ds_store_b32 are
  // bank-conflict-free with WSTR = 1028.
#pragma unroll
  for (int it = 0; it < (QDIM * 16) / 4 / 256; ++it) {
    const int idx = it * 256 + tid;
    const int f4  = idx & 3;          // which 4-wide chunk of the 16 columns
    const int k   = idx >> 2;
    const float4 v = *(const float4*)&W[(size_t)k * CDIM + n0 + f4 * 4];
    ldsB[(f4 * 4 + 0) * WSTR + k] = v.x;
    ldsB[(f4 * 4 + 1) * WSTR + k] = v.y;
    ldsB[(f4 * 4 + 2) * WSTR + k] = v.z;
    ldsB[(f4 * 4 + 3) * WSTR + k] = v.w;
  }
  __syncthreads();

  const float* __restrict__ xrow = X + (size_t)(m0 + wv * 16 + l16) * QDIM + kg;
  const float* brow = &ldsB[l16 * WSTR + kg];

  v8f acc = {};
#pragma unroll 4
  for (int kb = 0; kb < QDIM; kb += 4) {
    v2f a = *(const v2f*)(xrow + kb);
    v2f b = *(const v2f*)(brow + kb);
    acc = __builtin_amdgcn_wmma_f32_16x16x4_f32(
        /*neg_a=*/false, a, /*neg_b=*/false, b,
        /*c_mod=*/(short)0, acc, /*reuse_a=*/false, /*reuse_b=*/false);
  }

#pragma unroll
  for (int r = 0; r < 8; ++r) {
    const int row = m0 + wv * 16 + r + 8 * half;
    Y[(size_t)row * CDIM + n0 + l16] = acc[r];
  }
}

// ---------------------------------------------------------------------------
// Stage 2: scores[b,q,k] = sum_c tanh(q2[b,q,c] + k2[b,k,c]) * w[c] + bias
// 256 threads, 32q x 32k tile per block. q2/k2 tiles + w staged in LDS.
// Each thread: 1 k-column, 4 q-rows (q = (tid>>5) + 8*i).
// ---------------------------------------------------------------------------
#define QT  32
#define KT  32
#define STR 260   // padded row stride (floats): breaks stride-256 bank pattern

__global__ __launch_bounds__(256)
void additive_attn_scores(const float* __restrict__ q2,
                          const float* __restrict__ k2,
                          const float* __restrict__ w,
                          const float* __restrict__ bptr,
                          float* __restrict__ out) {
  __shared__ __align__(16) float ldsQ[QT * STR];
  __shared__ __align__(16) float ldsK[KT * STR];
  __shared__ __align__(16) float ldsW[CDIM];

  const int tid = threadIdx.x;
  const int b   = blockIdx.z;
  const int q0  = blockIdx.y * QT;
  const int k0  = blockIdx.x * KT;

  const float* qbase = q2 + ((size_t)b * TQ + q0) * CDIM;
  const float* kbase = k2 + ((size_t)b * TK + k0) * CDIM;

  // Cooperative tile load: 32 rows x 64 float4 = 2048 float4 per tile,
  // 8 per thread, fully coalesced on the global side.
#pragma unroll
  for (int it = 0; it < QT * (CDIM / 4) / 256; ++it) {
    const int idx = it * 256 + tid;
    const int row = idx >> 6;        // /64
    const int c4  = idx & 63;
    float4 v = ((const float4*)(qbase + (size_t)row * CDIM))[c4];
    *(float4*)&ldsQ[row * STR + c4 * 4] = v;
    float4 u = ((const float4*)(kbase + (size_t)row * CDIM))[c4];
    *(float4*)&ldsK[row * STR + c4 * 4] = u;
  }
  if (tid < CDIM / 4) {
    ((float4*)ldsW)[tid] = ((const float4*)w)[tid];
  }
  __syncthreads();

  const int kl = tid & 31;   // k within tile
  const int qg = tid >> 5;   // 0..7, q rows are qg + 8*i

  float s0 = 0.f, s1 = 0.f, s2 = 0.f, s3 = 0.f;

  for (int c = 0; c < CDIM; c += 4) {
    const float4 kv = *(const float4*)&ldsK[kl * STR + c];
    const float4 wv = *(const float4*)&ldsW[c];            // broadcast

    float4 qv;
    qv = *(const float4*)&ldsQ[(qg + 0) * STR + c];        // broadcast
    s0 += ftanh(qv.x + kv.x) * wv.x + ftanh(qv.y + kv.y) * wv.y
        + ftanh(qv.z + kv.z) * wv.z + ftanh(qv.w + kv.w) * wv.w;

    qv = *(const float4*)&ldsQ[(qg + 8) * STR + c];
    s1 += ftanh(qv.x + kv.x) * wv.x + ftanh(qv.y + kv.y) * wv.y
        + ftanh(qv.z + kv.z) * wv.z + ftanh(qv.w + kv.w) * wv.w;

    qv = *(const float4*)&ldsQ[(qg + 16) * STR + c];
    s2 += ftanh(qv.x + kv.x) * wv.x + ftanh(qv.y + kv.y) * wv.y
        + ftanh(qv.z + kv.z) * wv.z + ftanh(qv.w + kv.w) * wv.w;

    qv = *(const float4*)&ldsQ[(qg + 24) * STR + c];
    s3 += ftanh(qv.x + kv.x) * wv.x + ftanh(qv.y + kv.y) * wv.y
        + ftanh(qv.z + kv.z) * wv.z + ftanh(qv.w + kv.w) * wv.w;
  }

  const float bias = *bptr;
  float* orow = out + ((size_t)b * TQ) * TK;
  orow[(size_t)(q0 + qg +  0) * TK + k0 + kl] = s0 + bias;
  orow[(size_t)(q0 + qg +  8) * TK + k0 + kl] = s1 + bias;
  orow[(size_t)(q0 + qg + 16) * TK + k0 + kl] = s2 + bias;
  orow[(size_t)(q0 + qg + 24) * TK + k0 + kl] = s3 + bias;
}

// ---------------------------------------------------------------------------
extern "C" void kernel_launch(void* const* d_in, const int* in_sizes, int n_in,
                              void* d_out, int out_size, void* d_ws, size_t ws_size,
                              hipStream_t stream) {
  const float* query  = (const float*)d_in[0];   // (B, Tq, QDIM)
  const float* key    = (const float*)d_in[1];   // (B, Tk, QDIM)
  const float* Wq     = (const float*)d_in[2];   // (QDIM, CDIM)
  const float* Wk     = (const float*)d_in[3];   // (QDIM, CDIM)
  const float* w_attn = (const float*)d_in[4];   // (CDIM,)
  const float* b_attn = (const float*)d_in[5];   // scalar
  float* out = (float*)d_out;                    // (B, Tq, Tk)

  // Workspace: q2ctx then k2ctx, each (B*Tq, CDIM) f32 = 2 MB.
  float* q2 = (float*)d_ws;
  float* k2 = q2 + (size_t)BATCH * TQ * CDIM;

  // Stage 1: two WMMA GEMMs, (2048 x 1024) @ (1024 x 256).
  dim3 g1((BATCH * TQ) / 128, CDIM / 16);
  proj_gemm_wmma<<<g1, 256, 0, stream>>>(query, Wq, q2);
  proj_gemm_wmma<<<g1, 256, 0, stream>>>(key,   Wk, k2);

  // Stage 2: tiled tanh-reduce.
  dim3 g2(TK / KT, TQ / QT, BATCH);
  additive_attn_scores<<<g2, 256, 0, stream>>>(q2, k2, w_attn, b_attn, out);
}